// KNNClassifierLayer2_71966472011990
// MI455X (gfx1250) — compile-verified
//
#include <hip/hip_runtime.h>

typedef __attribute__((ext_vector_type(2))) float v2f;
typedef __attribute__((ext_vector_type(4))) float v4f;
typedef __attribute__((ext_vector_type(8))) float v8f;

#define KNN_K 5
#define NUM_CLASSES 6
#define DIM 32
#define TILE 16
#define WAVES 8
#define FLT_BIG 3.402823466e38f

// ---------------------------------------------------------------------------
// Kernel 1: t2[n] = ||X_train[n]||^2   (tiny, L2-resident)
// ---------------------------------------------------------------------------
__global__ void knn_t2_kernel(const float* __restrict__ Xt,
                              float* __restrict__ t2, int n) {
  int i = blockIdx.x * blockDim.x + threadIdx.x;
  if (i < n) {
    const float* p = Xt + i * DIM;
    float s = 0.0f;
#pragma unroll
    for (int j = 0; j < DIM; j += 4) {
      v4f v = *(const v4f*)(p + j);
      s += v[0] * v[0] + v[1] * v[1] + v[2] * v[2] + v[3] * v[3];
    }
    t2[i] = s;
  }
}

__device__ __forceinline__ void sort_step(float& a, int& ai, float& b, int& bi) {
  // ensure a <= b (ties keep earlier entry => smaller index first)
  if (b < a) {
    float td = a; a = b; b = td;
    int   ti = ai; ai = bi; bi = ti;
  }
}

struct ChunkBuf {
  v2f a[8];   // A-operands (train tile, 16xK slices)
  float tq;   // t2 of train row (lane%16) -- used by low lane-half only
};

// ---------------------------------------------------------------------------
// Kernel 2: fused WMMA-f32 GEMM + per-query top-5 + majority vote.
// Score = t2[m] - 2*dot(t_m, x_q) (x^2 is a per-query constant, same order).
// t2 is folded into the GEMM as a trailing WMMA: A-extra[m]=t2[m] on lane-
// half 0, B-extra=1, so the accumulator IS the ranking key.
// Pipeline distance 2 with UNCONDITIONAL clamped loads: each stage is one
// straight-line block (clause of 9 loads -> 9 WMMAs -> select), so the
// waitcnt pass can keep the newest clauses outstanding.
// ---------------------------------------------------------------------------
__launch_bounds__(WAVES * 32)
__global__ void knn_topk_vote_kernel(const float* __restrict__ x,
                                     const float* __restrict__ Xt,
                                     const int* __restrict__ y,
                                     const float* __restrict__ t2,
                                     float* __restrict__ out,
                                     int ntrain) {
  __shared__ float s_d[WAVES][2][TILE][KNN_K];
  __shared__ int   s_i[WAVES][2][TILE][KNN_K];

  const int tid   = threadIdx.x;
  const int wave  = tid >> 5;
  const int lane  = tid & 31;
  const int n     = lane & 15;   // query column / train row within tile
  const int h     = lane >> 4;   // lane half (K split)
  const int qbase = blockIdx.x * TILE;

  // ---- query tile B operands (scaled by -2), resident for whole loop
  v2f bq[8];
  {
    const float* xp = x + (qbase + n) * DIM + 2 * h;
#pragma unroll
    for (int t = 0; t < 8; ++t) {
      v2f v = *(const v2f*)(xp + 4 * t);
      bq[t] = v * (-2.0f);
    }
  }
  // B operand for the t2-fold WMMA: column of ones on K=0 (low half only)
  v2f bone;
  bone[0] = (h == 0) ? 1.0f : 0.0f;
  bone[1] = 0.0f;
  const float hmask = (h == 0) ? 1.0f : 0.0f;

  // ---- per-lane sorted top-5 (score, train index)
  float bd[KNN_K];
  int   bi[KNN_K];
#pragma unroll
  for (int j = 0; j < KNN_K; ++j) { bd[j] = FLT_BIG; bi[j] = 0; }

  const int nchunks = ntrain / TILE;  // 3125 for N=50000
  const int clast   = nchunks - 1;

  // Unconditional load with clamped chunk index (always a valid address;
  // clamped results are never consumed -- keeps stages branch-free).
  auto load_chunk = [&](int c, ChunkBuf& buf) {
    int cc = (c < clast) ? c : clast;
    const float* ap = Xt + (cc * TILE + n) * DIM + 2 * h;
#pragma unroll
    for (int t = 0; t < 8; ++t) buf.a[t] = *(const v2f*)(ap + 4 * t);
    buf.tq = t2[cc * TILE + n];
  };

  auto process_chunk = [&](int c, const ChunkBuf& buf) {
    v8f acc = __builtin_amdgcn_wmma_f32_16x16x4_f32(
        false, buf.a[0], false, bq[0], (short)0, v8f{}, false, false);
#pragma unroll
    for (int t = 1; t < 8; ++t)
      acc = __builtin_amdgcn_wmma_f32_16x16x4_f32(false, buf.a[t], false, bq[t],
                                                  (short)0, acc, false, false);
    // trailing t2 fold (depends on the last load of the clause, so it sits
    // at the end of the wait ladder)
    v2f at2;
    at2[0] = buf.tq * hmask;
    at2[1] = 0.0f;
    acc = __builtin_amdgcn_wmma_f32_16x16x4_f32(false, at2, false, bone,
                                                (short)0, acc, false, false);

    float mn = fminf(acc[0], acc[1]);
    mn = fminf(mn, fminf(acc[2], acc[3]));
    mn = fminf(mn, fminf(acc[4], acc[5]));
    mn = fminf(mn, fminf(acc[6], acc[7]));

    if (mn < bd[KNN_K - 1]) {  // exec-skip: rare after warmup
      const int ibase = c * TILE + 8 * h;
#pragma unroll
      for (int v = 0; v < 8; ++v) {
        float cv = acc[v];
        if (cv < bd[KNN_K - 1]) {
          bd[KNN_K - 1] = cv;
          bi[KNN_K - 1] = ibase + v;
#pragma unroll
          for (int j = KNN_K - 1; j > 0; --j)
            sort_step(bd[j - 1], bi[j - 1], bd[j], bi[j]);
        }
      }
    }
  };

  // ---- pipeline distance 2, 3-buffer rotation (wave-uniform control flow)
  {
    ChunkBuf bufA, bufB, bufC;
    int c = wave;
    if (c < nchunks) {
      load_chunk(c, bufA);
      load_chunk(c + WAVES, bufB);
      while (true) {
        load_chunk(c + 2 * WAVES, bufC);
        process_chunk(c, bufA);
        c += WAVES;
        if (c >= nchunks) break;

        load_chunk(c + 2 * WAVES, bufA);
        process_chunk(c, bufB);
        c += WAVES;
        if (c >= nchunks) break;

        load_chunk(c + 2 * WAVES, bufB);
        process_chunk(c, bufC);
        c += WAVES;
        if (c >= nchunks) break;
      }
    }
  }

  // ---- spill per-lane top-5 to LDS
#pragma unroll
  for (int j = 0; j < KNN_K; ++j) {
    s_d[wave][h][n][j] = bd[j];
    s_i[wave][h][n][j] = bi[j];
  }
  __syncthreads();

  // ---- one lane per query: merge WAVES x 2 halves x 5, then vote
  if (tid < TILE) {
    float fd[KNN_K];
    int   fi[KNN_K];
#pragma unroll
    for (int j = 0; j < KNN_K; ++j) { fd[j] = FLT_BIG; fi[j] = 0; }

    for (int w = 0; w < WAVES; ++w)
      for (int hh = 0; hh < 2; ++hh)
#pragma unroll
        for (int j = 0; j < KNN_K; ++j) {
          float cv = s_d[w][hh][tid][j];
          if (cv < fd[KNN_K - 1]) {
            fd[KNN_K - 1] = cv;
            fi[KNN_K - 1] = s_i[w][hh][tid][j];
#pragma unroll
            for (int jj = KNN_K - 1; jj > 0; --jj)
              sort_step(fd[jj - 1], fi[jj - 1], fd[jj], fi[jj]);
          }
        }

    // majority vote; argmax ties break to smallest class (strict >)
    int cnt[NUM_CLASSES];
#pragma unroll
    for (int cc = 0; cc < NUM_CLASSES; ++cc) cnt[cc] = 0;
#pragma unroll
    for (int j = 0; j < KNN_K; ++j) {
      int lbl = y[fi[j]];
      cnt[lbl] += 1;
    }
    int best = 0, bc = cnt[0];
#pragma unroll
    for (int cc = 1; cc < NUM_CLASSES; ++cc)
      if (cnt[cc] > bc) { bc = cnt[cc]; best = cc; }

    float* op = out + (qbase + tid) * NUM_CLASSES;
#pragma unroll
    for (int cc = 0; cc < NUM_CLASSES; ++cc)
      op[cc] = (cc == best) ? 1.0f : 0.0f;
  }
}

// ---------------------------------------------------------------------------
extern "C" void kernel_launch(void* const* d_in, const int* in_sizes, int n_in,
                              void* d_out, int out_size, void* d_ws, size_t ws_size,
                              hipStream_t stream) {
  const float* x  = (const float*)d_in[0];   // [B, 32]
  const float* Xt = (const float*)d_in[1];   // [N, 32]
  const int*   y  = (const int*)d_in[2];     // [N]
  float* out = (float*)d_out;                // [B, 6]
  float* t2  = (float*)d_ws;                 // [N] scratch

  const int N = in_sizes[2];
  const int B = in_sizes[0] / DIM;

  knn_t2_kernel<<<(N + 255) / 256, 256, 0, stream>>>(Xt, t2, N);
  knn_topk_vote_kernel<<<B / TILE, WAVES * 32, 0, stream>>>(x, Xt, y, t2, out, N);
}